// LoSS2Dv2_72018011619546
// MI455X (gfx1250) — compile-verified
//
#include <hip/hip_runtime.h>
#include <math.h>

#define DM 96
#define DI 192
#define NK 4
#define RNK 6
#define NB 4
#define HH 128
#define WW 128
#define LL 16384

typedef __attribute__((ext_vector_type(2))) float v2f;
typedef __attribute__((ext_vector_type(8))) float v8f;
typedef __attribute__((ext_vector_type(4))) unsigned u32x4;
typedef __attribute__((ext_vector_type(4))) int i32x4;
typedef __attribute__((ext_vector_type(8))) int i32x8;

#if __has_builtin(__builtin_amdgcn_tensor_load_to_lds)
#define HAVE_TDM 1
#if __has_include(<hip/amd_detail/amd_gfx1250_TDM.h>)
#define TDM_6ARG 1  // amdgpu-toolchain (clang-23 / therock headers): 6-arg builtin
#endif
#endif

static __device__ __forceinline__ v8f wmma_f32_k4(v2f a, v2f b, v8f c) {
  // D = A(16x4,f32) * B(4x16,f32) + C(16x16,f32)
  return __builtin_amdgcn_wmma_f32_16x16x4_f32(false, a, false, b, (short)0, c,
                                               false, false);
}

static __device__ __forceinline__ float softplusf(float x) {
  return (x > 20.f) ? x : log1pf(__expf(x));
}

static __device__ __forceinline__ float gelu_tanh(float x) {
  float x3 = x * x * x;
  return 0.5f * x * (1.f + tanhf(0.7978845608028654f * (x + 0.044715f * x3)));
}

static __device__ __forceinline__ void wait_tensor0() {
#if __has_builtin(__builtin_amdgcn_s_wait_tensorcnt)
  __builtin_amdgcn_s_wait_tensorcnt(0);
#else
  asm volatile("s_wait_tensorcnt 0x0" ::: "memory");
#endif
}

#ifdef HAVE_TDM
// TDM 2-D tile load: `rows` x 128 fp32 tile from a row-major tensor with row
// stride LL elements, into LDS at `panel`.  D# per CDNA5 ISA ch.8:
//   group0: count=1 | lds_addr[63:32] | global_addr[120:64] | type=2[127:126]
//   group1: data_size=4B[17:16], tensor_dim0=LL[79:48], tensor_dim1=rows[111:80],
//           tile_dim0=128[127:112], tile_dim1=rows[143:128],
//           tensor_dim0_stride=LL[207:160]
//   groups 2/3: zero (2-D tensor)
static __device__ __forceinline__ void tdm_load_panel(const float* g,
                                                      float* panel, int rows) {
  const unsigned long long ga = (unsigned long long)(uintptr_t)g;
  // generic LDS pointer: low 32 bits of the flat address are the LDS offset
  const unsigned ldsoff = (unsigned)(uintptr_t)panel;
  u32x4 g0;
  g0.x = 1u;                                   // count=1, no gather/restore
  g0.y = ldsoff;                               // lds_addr
  g0.z = (unsigned)ga;                         // global_addr[31:0]
  g0.w = (unsigned)((ga >> 32) & 0x01FFFFFFu)  // global_addr[56:32]
         | 0x80000000u;                        // type=2 ("image")
  i32x8 g1;
  g1[0] = 0x00020000;                 // workgroup_mask=0, data_size=2 (4 bytes)
  g1[1] = (int)(((unsigned)LL & 0xFFFFu) << 16);  // tensor_dim0[15:0] @ bit48
  g1[2] = (int)(((unsigned)LL >> 16) & 0xFFFFu)   // tensor_dim0[31:16]
          | (int)((unsigned)rows << 16);          // tensor_dim1[15:0] @ bit80
  g1[3] = (int)(128u << 16);          // tensor_dim1[31:16]=0 | tile_dim0=128
  g1[4] = rows;                       // tile_dim1=rows, tile_dim2=0
  g1[5] = LL;                         // tensor_dim0_stride[31:0]
  g1[6] = 0;                          // stride0[47:32]=0, stride1[15:0]=0
  g1[7] = 0;
  i32x4 z4 = {0, 0, 0, 0};
#ifdef TDM_6ARG
  i32x8 z8 = {0, 0, 0, 0, 0, 0, 0, 0};
  __builtin_amdgcn_tensor_load_to_lds(g0, g1, z4, z4, z8, 0);
#else
  __builtin_amdgcn_tensor_load_to_lds(g0, g1, z4, z4, 0);
#endif
}
#endif

// ---------------------------------------------------------------------------
// K1: in_proj  Y[b,m,n] = sum_c W[m,c] * X[b,c,n]   (M=192, K=96, N=L per b)
// One workgroup owns ALL of M for a 128-column panel; the 96x128 fp32 X panel
// (48KB) is staged in LDS once by the Tensor Data Mover and reused by all 12
// m-tiles (cuts B-operand global traffic 12x). 8 waves x 3 wave-tiles of
// 16(M)x64(N), f32 WMMA, K-step 4.
// ---------------------------------------------------------------------------
__global__ __launch_bounds__(256) void k_inproj(const float* __restrict__ W,
                                                const float* __restrict__ X,
                                                float* __restrict__ Y) {
  __shared__ float panel[DM * 128];  // 48 KB
  const int lane = threadIdx.x & 31;
  const int wave = threadIdx.x >> 5;
  const int hl = lane >> 4;
  const int l15 = lane & 15;
  const int b = blockIdx.y;
  const int n0 = blockIdx.x * 128;
  const float* Xb = X + (size_t)b * DM * LL;
#ifdef HAVE_TDM
  if (threadIdx.x < 32) {  // single wave issues one DMA for the whole panel
    tdm_load_panel(Xb + n0, panel, DM);
    wait_tensor0();
  }
#else
  for (int i = threadIdx.x; i < DM * 128; i += 256)
    panel[i] = Xb[(size_t)(i >> 7) * LL + n0 + (i & 127)];
#endif
  __syncthreads();

  const int nh = (wave & 1) * 64;  // column half within the panel
  const int mg = wave >> 1;        // m-tile group 0..3
  float* Yb = Y + (size_t)b * DI * LL;
  for (int j = 0; j < 3; ++j) {
    const int m0 = mg * 16 + j * 64;  // covers all 12 m-tiles across waves
    const float* wrow = W + (size_t)(m0 + l15) * DM;
    v8f c0 = {}; v8f c1 = {}; v8f c2 = {}; v8f c3 = {};
    for (int k0 = 0; k0 < DM; k0 += 4) {
      const int ka = k0 + 2 * hl;  // lanes 0-15: K=k0,k0+1; lanes 16-31: +2,+3
      v2f a; a.x = wrow[ka]; a.y = wrow[ka + 1];
      const float* lp0 = &panel[ka * 128 + nh + l15];
      const float* lp1 = lp0 + 128;
      v2f b0; b0.x = lp0[0];  b0.y = lp1[0];
      v2f b1; b1.x = lp0[16]; b1.y = lp1[16];
      v2f b2; b2.x = lp0[32]; b2.y = lp1[32];
      v2f b3; b3.x = lp0[48]; b3.y = lp1[48];
      c0 = wmma_f32_k4(a, b0, c0);
      c1 = wmma_f32_k4(a, b1, c1);
      c2 = wmma_f32_k4(a, b2, c2);
      c3 = wmma_f32_k4(a, b3, c3);
    }
#pragma unroll
    for (int r = 0; r < 8; ++r) {
      const int row = m0 + r + 8 * hl;
      float* yr = Yb + (size_t)row * LL + n0 + nh + l15;
      yr[0]  = c0[r];
      yr[16] = c1[r];
      yr[32] = c2[r];
      yr[48] = c3[r];
    }
  }
}

// ---------------------------------------------------------------------------
// K2: depthwise 3x3 conv (SAME) + bias + SiLU.  One thread per output element.
// ---------------------------------------------------------------------------
__global__ __launch_bounds__(256) void k_dwconv(const float* __restrict__ X,
                                                const float* __restrict__ Wc,
                                                const float* __restrict__ Bc,
                                                float* __restrict__ Y) {
  const int idx = blockIdx.x * 256 + threadIdx.x;
  const int w = idx & (WW - 1);
  const int h = (idx >> 7) & (HH - 1);
  const int d = (idx >> 14) % DI;
  const int b = idx / (LL * DI);
  const float* xp = X + (size_t)(b * DI + d) * LL;
  const float* wp = Wc + d * 9;
  float acc = Bc[d];
#pragma unroll
  for (int di = -1; di <= 1; ++di) {
    const int hh = h + di;
    if (hh < 0 || hh >= HH) continue;
#pragma unroll
    for (int dj = -1; dj <= 1; ++dj) {
      const int ww2 = w + dj;
      if (ww2 < 0 || ww2 >= WW) continue;
      acc += wp[(di + 1) * 3 + (dj + 1)] * xp[hh * WW + ww2];
    }
  }
  acc = acc / (1.f + __expf(-acc));  // SiLU
  Y[(size_t)(b * DI + d) * LL + h * WW + w] = acc;
}

// ---------------------------------------------------------------------------
// K3: x_dbl in SPATIAL order (coalesced):
//   XD[b,k,c,p] = sum_d XPW[k,c,d] * XC[b,d,p]   (c = 0..7)
// The scan kernel gathers these 8 rows at p = scan_ids[k,l].
// ---------------------------------------------------------------------------
__global__ __launch_bounds__(256) void k_xdbl(const float* __restrict__ XC,
                                              const float* __restrict__ XPW,
                                              float* __restrict__ XD) {
  const int idx = blockIdx.x * 256 + threadIdx.x;
  const int p = idx & (LL - 1);
  const int k = (idx >> 14) & 3;
  const int b = idx >> 16;
  const float* xc = XC + (size_t)b * DI * LL + p;
  const float* wp = XPW + k * 8 * DI;
  float acc[8] = {0.f, 0.f, 0.f, 0.f, 0.f, 0.f, 0.f, 0.f};
  for (int d = 0; d < DI; ++d) {
    const float v = xc[(size_t)d * LL];
#pragma unroll
    for (int c = 0; c < 8; ++c) acc[c] += wp[c * DI + d] * v;
  }
  float* o = XD + ((size_t)(b * NK + k) * 8) * LL + p;
#pragma unroll
  for (int c = 0; c < 8; ++c) o[(size_t)c * LL] = acc[c];
}

// ---------------------------------------------------------------------------
// K4: selective scan (D_STATE=1 -> scalar recurrence), one workgroup per
// (b,d). Per direction: chunked prefix (64 elems/thread), block combine,
// replay, conflict-free scatter-accumulate into a 64KB LDS row (p is a
// permutation within a direction; directions separated by barriers). GELU
// fused on writeback. Deterministic: no atomics.
// ---------------------------------------------------------------------------
__global__ __launch_bounds__(256) void k_scan(const float* __restrict__ XC,
                                              const float* __restrict__ XD,
                                              const int* __restrict__ SID,
                                              const float* __restrict__ DTW,
                                              const float* __restrict__ DTB,
                                              const float* __restrict__ ALOG,
                                              const float* __restrict__ DS,
                                              float* __restrict__ YS) {
  __shared__ float yrow[LL];  // 64 KB (CDNA5: up to 320KB LDS per workgroup)
  __shared__ float sA[256];
  __shared__ float sB[256];
  __shared__ float sH[256];
  const int t = threadIdx.x;
  const int d = blockIdx.x % DI;
  const int b = blockIdx.x / DI;
  for (int i = t; i < LL; i += 256) yrow[i] = 0.f;
  __syncthreads();
  const float* xcb = XC + (size_t)(b * DI + d) * LL;
  for (int k = 0; k < NK; ++k) {
    const int kd = k * DI + d;
    const float Akd = -__expf(ALOG[kd]);  // A = -exp(A_log), D_STATE==1
    const float Dkd = DS[kd];
    const float bias = DTB[kd];
    float wr[RNK];
#pragma unroll
    for (int r = 0; r < RNK; ++r) wr[r] = DTW[kd * RNK + r];
    const float* xd = XD + ((size_t)(b * NK + k) * 8) * LL;
    const int* sid = SID + k * LL;
    const int l0 = t * 64;
    // ---- pass 1: per-chunk aggregate (Aprod, h from 0) ----
    float Ap = 1.f, hc = 0.f;
    for (int i = 0; i < 64; ++i) {
      const int p = sid[l0 + i];
      float dt = bias;
#pragma unroll
      for (int r = 0; r < RNK; ++r) dt += wr[r] * xd[(size_t)r * LL + p];
      const float delta = softplusf(dt);
      const float a = __expf(delta * Akd);
      const float u = xcb[p];
      const float bu = delta * xd[(size_t)6 * LL + p] * u;
      hc = a * hc + bu;
      Ap *= a;
    }
    sA[t] = Ap;
    sB[t] = hc;
    __syncthreads();
    // ---- block combine: exclusive prefix h for each chunk ----
    if (t == 0) {
      float hrun = 0.f;
      for (int q = 0; q < 256; ++q) {
        sH[q] = hrun;
        hrun = sA[q] * hrun + sB[q];
      }
    }
    __syncthreads();
    // ---- pass 2: replay with prefix, emit y, scatter into LDS row ----
    float h = sH[t];
    for (int i = 0; i < 64; ++i) {
      const int p = sid[l0 + i];
      float dt = bias;
#pragma unroll
      for (int r = 0; r < RNK; ++r) dt += wr[r] * xd[(size_t)r * LL + p];
      const float delta = softplusf(dt);
      const float a = __expf(delta * Akd);
      const float u = xcb[p];
      const float bu = delta * xd[(size_t)6 * LL + p] * u;
      h = a * h + bu;
      const float y = xd[(size_t)7 * LL + p] * h + Dkd * u;
      yrow[p] += y;  // p unique within this direction -> no conflict
    }
    __syncthreads();
  }
  float* yo = YS + (size_t)(b * DI + d) * LL;
  for (int i = t; i < LL; i += 256) yo[i] = gelu_tanh(yrow[i]);
}

// ---------------------------------------------------------------------------
// K5: out_proj  O[b,o,n] = sum_c W[o,c] * Y[b,c,n]  (M=96, K=192, N=L per b)
// Y already GELU'd (fused into K4). 192x128 fp32 panel (96KB) staged by TDM,
// reused by all 6 m-tiles. 6 waves x 2 wave-tiles of 16x64.
// ---------------------------------------------------------------------------
__global__ __launch_bounds__(192) void k_outproj(const float* __restrict__ W,
                                                 const float* __restrict__ X,
                                                 float* __restrict__ Y) {
  __shared__ float panel[DI * 128];  // 96 KB
  const int lane = threadIdx.x & 31;
  const int wave = threadIdx.x >> 5;
  const int hl = lane >> 4;
  const int l15 = lane & 15;
  const int b = blockIdx.y;
  const int n0 = blockIdx.x * 128;
  const float* Xb = X + (size_t)b * DI * LL;
#ifdef HAVE_TDM
  if (threadIdx.x < 32) {
    tdm_load_panel(Xb + n0, panel, DI);
    wait_tensor0();
  }
#else
  for (int i = threadIdx.x; i < DI * 128; i += 192)
    panel[i] = Xb[(size_t)(i >> 7) * LL + n0 + (i & 127)];
#endif
  __syncthreads();

  const int nh = (wave % 2) * 64;
  const int mg = wave / 2;  // 0..2
  float* Yb = Y + (size_t)b * DM * LL;
  for (int j = 0; j < 2; ++j) {
    const int m0 = mg * 16 + j * 48;  // covers all 6 m-tiles across waves
    const float* wrow = W + (size_t)(m0 + l15) * DI;
    v8f c0 = {}; v8f c1 = {}; v8f c2 = {}; v8f c3 = {};
    for (int k0 = 0; k0 < DI; k0 += 4) {
      const int ka = k0 + 2 * hl;
      v2f a; a.x = wrow[ka]; a.y = wrow[ka + 1];
      const float* lp0 = &panel[ka * 128 + nh + l15];
      const float* lp1 = lp0 + 128;
      v2f b0; b0.x = lp0[0];  b0.y = lp1[0];
      v2f b1; b1.x = lp0[16]; b1.y = lp1[16];
      v2f b2; b2.x = lp0[32]; b2.y = lp1[32];
      v2f b3; b3.x = lp0[48]; b3.y = lp1[48];
      c0 = wmma_f32_k4(a, b0, c0);
      c1 = wmma_f32_k4(a, b1, c1);
      c2 = wmma_f32_k4(a, b2, c2);
      c3 = wmma_f32_k4(a, b3, c3);
    }
#pragma unroll
    for (int r = 0; r < 8; ++r) {
      const int row = m0 + r + 8 * hl;
      float* yr = Yb + (size_t)row * LL + n0 + nh + l15;
      yr[0]  = c0[r];
      yr[16] = c1[r];
      yr[32] = c2[r];
      yr[48] = c3[r];
    }
  }
}

// ---------------------------------------------------------------------------
extern "C" void kernel_launch(void* const* d_in, const int* in_sizes, int n_in,
                              void* d_out, int out_size, void* d_ws,
                              size_t ws_size, hipStream_t stream) {
  (void)in_sizes; (void)n_in; (void)out_size; (void)ws_size;
  const float* x          = (const float*)d_in[0];   // (B,96,H,W)
  const int*   scan_ids   = (const int*)  d_in[1];   // (K,L)
  /* d_in[2] inverse_ids: unused (scatter with scan_ids is equivalent) */
  const float* in_proj_w  = (const float*)d_in[3];   // (192,96)
  const float* conv_w     = (const float*)d_in[4];   // (192,1,3,3)
  const float* conv_b     = (const float*)d_in[5];   // (192,)
  const float* xpw        = (const float*)d_in[6];   // (K,8,192)
  const float* dtw        = (const float*)d_in[7];   // (K,192,6)
  const float* dtb        = (const float*)d_in[8];   // (K,192)
  const float* A_logs     = (const float*)d_in[9];   // (K*192,1)
  const float* Ds         = (const float*)d_in[10];  // (K*192,)
  const float* out_proj_w = (const float*)d_in[11];  // (96,192)
  float* out = (float*)d_out;                        // (B,96,H,W)

  // Workspace layout (ysum aliases out1: out1 is dead after K2)
  const size_t SZ_BDL = (size_t)NB * DI * LL * sizeof(float);  // 50,331,648 B
  char* ws = (char*)d_ws;
  float* out1  = (float*)ws;                 // K1 -> K2 input
  float* xconv = (float*)(ws + SZ_BDL);      // K2 output
  float* xdbl  = (float*)(ws + 2 * SZ_BDL);  // K3 output (8.4 MB)
  float* ysum  = out1;                       // K4 output (aliases out1)

  k_inproj <<<dim3(LL / 128, NB), 256, 0, stream>>>(in_proj_w, x, out1);
  k_dwconv <<<(NB * DI * LL) / 256, 256, 0, stream>>>(out1, conv_w, conv_b,
                                                      xconv);
  k_xdbl   <<<(NB * NK * LL) / 256, 256, 0, stream>>>(xconv, xpw, xdbl);
  k_scan   <<<NB * DI, 256, 0, stream>>>(xconv, xdbl, scan_ids, dtw, dtb,
                                         A_logs, Ds, ysum);
  k_outproj<<<dim3(LL / 128, NB), 192, 0, stream>>>(out_proj_w, ysum, out);
}